// ModifiedMamba_61942018343327
// MI455X (gfx1250) — compile-verified
//
#include <hip/hip_runtime.h>
#include <hip/hip_bf16.h>

// ---------------------------------------------------------------------------
// Bidirectional Mamba block for MI455X (gfx1250), fp32 end-to-end.
// GEMMs use V_WMMA_F32_16X16X4_F32 (wave32) with 4x4 / 2x4 register blocking.
// __launch_bounds__(256,1) on the blocked kernels lifts the default VGPR cap
// (round 2 showed accumulators spilling to scratch under the ~128-VGPR
// occupancy heuristic).
// ---------------------------------------------------------------------------

typedef __attribute__((ext_vector_type(2))) float v2f;
typedef __attribute__((ext_vector_type(8))) float v8f;

#define SEQ     1024
#define DMODEL  1024
#define DINNER  2048
#define NSTATE  16
#define DTRANK  64
#define XDW     96            // DT_RANK + 2*D_STATE
#define NBATCH  2
#define MROWS   (NBATCH * SEQ)   // 2048 rows (b,l) flattened

// Workspace layout (floats)
#define XZ_ELEMS    ((size_t)MROWS * 2 * DINNER)
#define XC_ELEMS    ((size_t)MROWS * DINNER)
#define XDBL_ELEMS  ((size_t)MROWS * XDW)
#define DT_ELEMS    ((size_t)MROWS * DINNER)
#define YP_ELEMS    ((size_t)MROWS * DINNER)

// ---------------------------------------------------------------------------
// Register-blocked fp32 WMMA inner loop.
// Lane layout (lane = lo + 16*hi):
//   A frag vgpr r : [M = lo][K = kblk + 2*hi + r]  (contiguous pair -> b64)
//   B frag vgpr r : [K = kblk + 2*hi + r][N = lo]  (Bt row-major: Bt[n][k])
//   C/D   vgpr r : [M = r + 8*hi][N = lo]
// aptr/bptr already include (row lo)*ld + 2*hi.
// ---------------------------------------------------------------------------
template <int MI, int NI, int K, int LDA, int LDB>
__device__ __forceinline__ void wmma_block(const float* __restrict__ aptr,
                                           const float* __restrict__ bptr,
                                           v8f acc[MI][NI]) {
#pragma unroll 2
  for (int k = 0; k < K; k += 4) {
    v2f af[MI], bf[NI];
#pragma unroll
    for (int i = 0; i < MI; ++i)
      af[i] = *(const v2f*)(aptr + (size_t)i * 16 * LDA + k);
#pragma unroll
    for (int j = 0; j < NI; ++j)
      bf[j] = *(const v2f*)(bptr + (size_t)j * 16 * LDB + k);
#pragma unroll
    for (int i = 0; i < MI; ++i)
#pragma unroll
      for (int j = 0; j < NI; ++j)
        acc[i][j] = __builtin_amdgcn_wmma_f32_16x16x4_f32(
            false, af[i], false, bf[j], (short)0, acc[i][j], false, false);
  }
}

// ---------------------------------------------------------------------------
// K1: xz[m][n] = hidden[m][:] . W_in[n][:]  -- 64x64 tile per wave (4x4 blk)
// ---------------------------------------------------------------------------
__global__ void __launch_bounds__(256, 1)
k_gemm_in(const float* __restrict__ A,   // [2048,1024]
          const float* __restrict__ Bt,  // W_in [4096,1024]
          float* __restrict__ C) {       // xz [2048,4096]
  const int lane = threadIdx.x & 31;
  const int lo = lane & 15, hi = lane >> 4;
  const int tile = blockIdx.x * (blockDim.x >> 5) + (threadIdx.x >> 5);
  const int ntn = (2 * DINNER) / 64;             // 64
  const int tn = tile % ntn, tm = tile / ntn;    // tm in [0,32)

  const float* aptr = A + (size_t)(tm * 64 + lo) * DMODEL + 2 * hi;
  const float* bptr = Bt + (size_t)(tn * 64 + lo) * DMODEL + 2 * hi;

  v8f acc[4][4] = {};
  wmma_block<4, 4, DMODEL, DMODEL, DMODEL>(aptr, bptr, acc);

#pragma unroll
  for (int i = 0; i < 4; ++i)
#pragma unroll
    for (int r = 0; r < 8; ++r) {
      float* crow = C + (size_t)(tm * 64 + i * 16 + 8 * hi + r) * (2 * DINNER) +
                    tn * 64 + lo;
#pragma unroll
      for (int j = 0; j < 4; ++j) crow[j * 16] = acc[i][j][r];
    }
}

// ---------------------------------------------------------------------------
// K2: depthwise causal conv1d (k=4) + bias + SiLU, both directions.
// Backward branch computed in reversed-time coordinates t (l = SEQ-1-t).
// ---------------------------------------------------------------------------
__global__ void k_conv_silu(const float* __restrict__ xz,
                            const float* __restrict__ cw_f, const float* __restrict__ cb_f,
                            const float* __restrict__ cw_b, const float* __restrict__ cb_b,
                            float* __restrict__ xc) {     // [dir][2048][2048]
  const int idx = blockIdx.x * blockDim.x + threadIdx.x;  // 2^23 threads
  const int d   = idx & (DINNER - 1);
  const int t   = (idx >> 11) & (SEQ - 1);
  const int b   = (idx >> 21) & 1;
  const int dir = idx >> 22;

  const float* w = (dir ? cw_b : cw_f) + d * 4;
  float acc = (dir ? cb_b : cb_f)[d];
#pragma unroll
  for (int j = 0; j < 4; ++j) {
    const int tt = t + j - 3;                   // causal: left zero-pad
    if (tt >= 0) {
      const int l = dir ? (SEQ - 1 - tt) : tt;  // backward reads reversed seq
      acc += w[j] * xz[(size_t)(b * SEQ + l) * (2 * DINNER) + d];
    }
  }
  const float s = acc / (1.0f + __expf(-acc));  // SiLU
  xc[(size_t)dir * XC_ELEMS + (size_t)(b * SEQ + t) * DINNER + d] = s;
}

// ---------------------------------------------------------------------------
// K3: x_dbl[m][n] = xc[m][:] . Wx[n][:]  (N=96 -> keep 16x16 tiles), per dir.
// ---------------------------------------------------------------------------
__global__ void k_gemm_xdbl(const float* __restrict__ xc,
                            const float* __restrict__ Wx_f,
                            const float* __restrict__ Wx_b,
                            float* __restrict__ xdbl) {
  const int dir = blockIdx.y;
  const float* Bt = dir ? Wx_b : Wx_f;           // [96, 2048]
  const int lane = threadIdx.x & 31;
  const int lo = lane & 15, hi = lane >> 4;
  const int tile = blockIdx.x * (blockDim.x >> 5) + (threadIdx.x >> 5);
  const int ntn = XDW / 16;                      // 6
  const int tn = tile % ntn, tm = tile / ntn;

  const float* aptr = xc + (size_t)dir * XC_ELEMS +
                      (size_t)(tm * 16 + lo) * DINNER + 2 * hi;
  const float* bptr = Bt + (size_t)(tn * 16 + lo) * DINNER + 2 * hi;

  v8f acc[1][1] = {};
  wmma_block<1, 1, DINNER, DINNER, DINNER>(aptr, bptr, acc);

  float* crow = xdbl + (size_t)dir * XDBL_ELEMS +
                (size_t)(tm * 16 + 8 * hi) * XDW + tn * 16 + lo;
#pragma unroll
  for (int r = 0; r < 8; ++r) crow[(size_t)r * XDW] = acc[0][0][r];
}

// ---------------------------------------------------------------------------
// K4: dt[m][n] = softplus(x_dbl[m][0:64] . Wdt[n][:] + bdt[n]), per dir.
// 64x64 tile per wave (4x4 blocking), K = 64.
// ---------------------------------------------------------------------------
__global__ void __launch_bounds__(256, 1)
k_gemm_dt(const float* __restrict__ xdbl,
          const float* __restrict__ Wdt_f,
          const float* __restrict__ Wdt_b,
          const float* __restrict__ bdt_f,
          const float* __restrict__ bdt_b,
          float* __restrict__ dtb) {
  const int dir = blockIdx.y;
  const float* Bt = dir ? Wdt_b : Wdt_f;          // [2048, 64]
  const float* bias = dir ? bdt_b : bdt_f;
  const int lane = threadIdx.x & 31;
  const int lo = lane & 15, hi = lane >> 4;
  const int tile = blockIdx.x * (blockDim.x >> 5) + (threadIdx.x >> 5);
  const int ntn = DINNER / 64;                    // 32
  const int tn = tile % ntn, tm = tile / ntn;     // tm in [0,32)

  const float* aptr = xdbl + (size_t)dir * XDBL_ELEMS +
                      (size_t)(tm * 64 + lo) * XDW + 2 * hi;
  const float* bptr = Bt + (size_t)(tn * 64 + lo) * DTRANK + 2 * hi;

  v8f acc[4][4] = {};
  wmma_block<4, 4, DTRANK, XDW, DTRANK>(aptr, bptr, acc);

  float bn[4];
#pragma unroll
  for (int j = 0; j < 4; ++j) bn[j] = bias[tn * 64 + j * 16 + lo];

#pragma unroll
  for (int i = 0; i < 4; ++i)
#pragma unroll
    for (int r = 0; r < 8; ++r) {
      float* crow = dtb + (size_t)dir * DT_ELEMS +
                    (size_t)(tm * 64 + i * 16 + 8 * hi + r) * DINNER +
                    tn * 64 + lo;
#pragma unroll
      for (int j = 0; j < 4; ++j) {
        float v = acc[i][j][r] + bn[j];
        v = (v > 20.0f) ? v : log1pf(__expf(v));  // softplus
        crow[j * 16] = v;
      }
    }
}

// ---------------------------------------------------------------------------
// K5: selective scan. One thread per (dir, b, d, n); 16 lanes share a channel.
// y[t,d] = (sum_n state[n]*C[t,n]) + x*D, then * silu(z).
// ---------------------------------------------------------------------------
__global__ void k_scan(const float* __restrict__ dtb,
                       const float* __restrict__ xc,
                       const float* __restrict__ xdbl,
                       const float* __restrict__ xz,
                       const float* __restrict__ Alog_f, const float* __restrict__ Alog_b,
                       const float* __restrict__ Df, const float* __restrict__ Db,
                       float* __restrict__ yp) {
  const int idx = blockIdx.x * blockDim.x + threadIdx.x;  // 2^17 threads
  const int n   = idx & 15;
  const int d   = (idx >> 4) & (DINNER - 1);
  const int b   = (idx >> 15) & 1;
  const int dir = idx >> 16;

  const float A  = -__expf((dir ? Alog_b : Alog_f)[d * NSTATE + n]);
  const float Dd = (dir ? Db : Df)[d];

  const float* dtp  = dtb  + (size_t)dir * DT_ELEMS   + (size_t)(b * SEQ) * DINNER + d;
  const float* xp   = xc   + (size_t)dir * XC_ELEMS   + (size_t)(b * SEQ) * DINNER + d;
  const float* bc   = xdbl + (size_t)dir * XDBL_ELEMS + (size_t)(b * SEQ) * XDW;
  const float* zrow = xz   + (size_t)(b * SEQ) * (2 * DINNER) + DINNER + d;
  float*       yo   = yp   + (size_t)dir * YP_ELEMS   + (size_t)(b * SEQ) * DINNER + d;

  float carry = 0.0f;
  for (int t = 0; t < SEQ; ++t) {
    const float dtv = dtp[(size_t)t * DINNER];
    const float xv  = xp[(size_t)t * DINNER];
    const float Bv  = bc[t * XDW + DTRANK + n];
    const float Cv  = bc[t * XDW + DTRANK + NSTATE + n];
    const float dA  = __expf(dtv * A);
    carry = dA * carry + dtv * Bv * xv;
    float pv = carry * Cv;
    pv += __shfl_xor(pv, 1, 16);
    pv += __shfl_xor(pv, 2, 16);
    pv += __shfl_xor(pv, 4, 16);
    pv += __shfl_xor(pv, 8, 16);
    if (n == 0) {
      const int l = dir ? (SEQ - 1 - t) : t;
      const float z  = zrow[(size_t)l * (2 * DINNER)];
      const float sz = z / (1.0f + __expf(-z));
      yo[(size_t)t * DINNER] = (pv + xv * Dd) * sz;
    }
  }
}

// ---------------------------------------------------------------------------
// K6: out[m][n] = (0.5*(y_f[b,l,:] + y_b[b,SEQ-1-l,:])) . W_out[n][:]
// 32x64 tile per wave (2x4 blocking); combine fused into A-fragment load.
// Row reversal is per-row, so each M-subtile carries its own y_b pointer.
// ---------------------------------------------------------------------------
__global__ void __launch_bounds__(256, 1)
k_gemm_out(const float* __restrict__ yp,
           const float* __restrict__ Wout,   // [1024, 2048]
           float* __restrict__ out) {        // [2048, 1024]
  const int lane = threadIdx.x & 31;
  const int lo = lane & 15, hi = lane >> 4;
  const int tile = blockIdx.x * (blockDim.x >> 5) + (threadIdx.x >> 5);
  const int ntn = DMODEL / 64;                    // 16
  const int tn = tile % ntn, tm = tile / ntn;     // tm in [0,64)

  const float* af_p[2];
  const float* ab_p[2];
#pragma unroll
  for (int i = 0; i < 2; ++i) {
    const int m = tm * 32 + i * 16 + lo;
    const int b = m >> 10, l = m & (SEQ - 1);
    af_p[i] = yp + (size_t)m * DINNER + 2 * hi;
    ab_p[i] = yp + YP_ELEMS +
              (size_t)(b * SEQ + (SEQ - 1 - l)) * DINNER + 2 * hi;
  }
  const float* bptr = Wout + (size_t)(tn * 64 + lo) * DINNER + 2 * hi;

  v8f acc[2][4] = {};
#pragma unroll 2
  for (int k = 0; k < DINNER; k += 4) {
    v2f af[2], bf[4];
#pragma unroll
    for (int i = 0; i < 2; ++i) {
      v2f f = *(const v2f*)(af_p[i] + k);
      v2f r = *(const v2f*)(ab_p[i] + k);
      af[i] = (f + r) * 0.5f;
    }
#pragma unroll
    for (int j = 0; j < 4; ++j)
      bf[j] = *(const v2f*)(bptr + (size_t)j * 16 * DINNER + k);
#pragma unroll
    for (int i = 0; i < 2; ++i)
#pragma unroll
      for (int j = 0; j < 4; ++j)
        acc[i][j] = __builtin_amdgcn_wmma_f32_16x16x4_f32(
            false, af[i], false, bf[j], (short)0, acc[i][j], false, false);
  }

#pragma unroll
  for (int i = 0; i < 2; ++i)
#pragma unroll
    for (int r = 0; r < 8; ++r) {
      float* crow = out + (size_t)(tm * 32 + i * 16 + 8 * hi + r) * DMODEL +
                    tn * 64 + lo;
#pragma unroll
      for (int j = 0; j < 4; ++j) crow[j * 16] = acc[i][j][r];
    }
}

// ---------------------------------------------------------------------------
extern "C" void kernel_launch(void* const* d_in, const int* in_sizes, int n_in,
                              void* d_out, int out_size, void* d_ws, size_t ws_size,
                              hipStream_t stream) {
  const float* hidden  = (const float*)d_in[0];
  const float* W_in    = (const float*)d_in[1];
  const float* conv_w  = (const float*)d_in[2];
  const float* conv_b  = (const float*)d_in[3];
  const float* Wx      = (const float*)d_in[4];
  const float* Wdt     = (const float*)d_in[5];
  const float* bdt     = (const float*)d_in[6];
  const float* A_log   = (const float*)d_in[7];
  const float* Dv      = (const float*)d_in[8];
  const float* conv_wb = (const float*)d_in[9];
  const float* conv_bb = (const float*)d_in[10];
  const float* Wx_b    = (const float*)d_in[11];
  const float* Wdt_b   = (const float*)d_in[12];
  const float* bdt_b   = (const float*)d_in[13];
  const float* A_b_log = (const float*)d_in[14];
  const float* D_b     = (const float*)d_in[15];
  const float* W_out   = (const float*)d_in[16];
  float* out = (float*)d_out;

  float* xz   = (float*)d_ws;                 // [2048][4096]
  float* xc   = xz   + XZ_ELEMS;              // 2 x [2048][2048]
  float* xdbl = xc   + 2 * XC_ELEMS;          // 2 x [2048][96]
  float* dtb  = xdbl + 2 * XDBL_ELEMS;        // 2 x [2048][2048]
  float* yp   = dtb  + 2 * DT_ELEMS;          // 2 x [2048][2048]

  // K1: input projection GEMM; 2048 tiles of 64x64, 8 waves/block
  k_gemm_in<<<dim3(256), dim3(256), 0, stream>>>(hidden, W_in, xz);

  // K2: conv + SiLU, both dirs (2^23 threads)
  k_conv_silu<<<dim3(32768), dim3(256), 0, stream>>>(xz, conv_w, conv_b,
                                                     conv_wb, conv_bb, xc);

  // K3: x_dbl GEMM (768 tiles/dir of 16x16)
  k_gemm_xdbl<<<dim3(96, 2), dim3(256), 0, stream>>>(xc, Wx, Wx_b, xdbl);

  // K4: dt GEMM + softplus (1024 tiles/dir of 64x64)
  k_gemm_dt<<<dim3(128, 2), dim3(256), 0, stream>>>(xdbl, Wdt, Wdt_b,
                                                    bdt, bdt_b, dtb);

  // K5: selective scan (131072 threads)
  k_scan<<<dim3(512), dim3(256), 0, stream>>>(dtb, xc, xdbl, xz,
                                              A_log, A_b_log, Dv, D_b, yp);

  // K6: combine + output GEMM (1024 tiles of 32x64)
  k_gemm_out<<<dim3(128), dim3(256), 0, stream>>>(yp, W_out, out);
}